// MPS_19834158973394
// MI455X (gfx1250) — compile-verified
//
#include <hip/hip_runtime.h>
#include <hip/hip_bf16.h>
#include <math.h>

// ---------------------------------------------------------------------------
// MPS chain contraction for MI455X (gfx1250, wave32, WMMA f32 16x16x4).
//
// Math:  S' = sum_{p,q} M_pq B_p^H S B_q  with M = L L^H (Cholesky)
//        =>  S' = sum_k K_k^H S K_k,  K_k = sum_q conj(L_qk) B_q
// QR avoided: M_v = Rinv^H (A_v^H A_v) Rinv, R = chol(sum_v A_v^H A_v).
// ---------------------------------------------------------------------------

typedef __attribute__((ext_vector_type(2))) float v2f;
typedef __attribute__((ext_vector_type(8))) float v8f;

__device__ __forceinline__ v8f vzero8() {
    v8f v = {0.f, 0.f, 0.f, 0.f, 0.f, 0.f, 0.f, 0.f};
    return v;
}

__device__ __forceinline__ v8f wmma4(v2f a, v2f b, v8f c) {
    // D = A(16x4 f32) * B(4x16 f32) + C(16x16 f32)
    return __builtin_amdgcn_wmma_f32_16x16x4_f32(
        false, a, false, b, (short)0, c, false, false);
}

struct cx { float re, im; };
__device__ __forceinline__ cx cmul(cx a, cx b) { return {a.re * b.re - a.im * b.im, a.re * b.im + a.im * b.re}; }
__device__ __forceinline__ cx cadd(cx a, cx b) { return {a.re + b.re, a.im + b.im}; }
__device__ __forceinline__ cx csub(cx a, cx b) { return {a.re - b.re, a.im - b.im}; }
__device__ __forceinline__ cx cconj(cx a) { return {a.re, -a.im}; }

// ---------------------------------------------------------------------------
// Kernel 1: Gram matrix G = A^H A over (4096 x 4) rows, Cholesky, and Rinv.
// One block of 256 threads; thread 0 does the 4x4 factorization.
// ---------------------------------------------------------------------------
__global__ void k_gram_rinv(const float* __restrict__ a_re,
                            const float* __restrict__ a_im,
                            float* __restrict__ rinv /* 32 floats */) {
    __shared__ float red[256][33];
    const int tid = threadIdx.x;
    float g[32];
    #pragma unroll
    for (int e = 0; e < 32; ++e) g[e] = 0.f;

    for (int rr = tid; rr < 4096; rr += 256) {
        float ar[4], ai[4];
        #pragma unroll
        for (int j = 0; j < 4; ++j) { ar[j] = a_re[rr * 4 + j]; ai[j] = a_im[rr * 4 + j]; }
        #pragma unroll
        for (int i = 0; i < 4; ++i)
            #pragma unroll
            for (int j = 0; j < 4; ++j) {
                g[(i * 4 + j) * 2 + 0] += ar[i] * ar[j] + ai[i] * ai[j];
                g[(i * 4 + j) * 2 + 1] += ar[i] * ai[j] - ai[i] * ar[j];
            }
    }
    #pragma unroll
    for (int e = 0; e < 32; ++e) red[tid][e] = g[e];
    __syncthreads();
    for (int s = 128; s > 0; s >>= 1) {
        if (tid < s)
            for (int e = 0; e < 32; ++e) red[tid][e] += red[tid + s][e];
        __syncthreads();
    }
    if (tid == 0) {
        cx G[4][4];
        for (int i = 0; i < 4; ++i)
            for (int j = 0; j < 4; ++j)
                G[i][j] = {red[0][(i * 4 + j) * 2], red[0][(i * 4 + j) * 2 + 1]};
        // Cholesky: G = Lc Lc^H (Lc lower)
        cx Lc[4][4] = {};
        for (int k = 0; k < 4; ++k) {
            float d = G[k][k].re;
            for (int m = 0; m < k; ++m) d -= Lc[k][m].re * Lc[k][m].re + Lc[k][m].im * Lc[k][m].im;
            d = sqrtf(fmaxf(d, 1e-30f));
            Lc[k][k] = {d, 0.f};
            float inv = 1.f / d;
            for (int i = k + 1; i < 4; ++i) {
                cx s = G[i][k];
                for (int m = 0; m < k; ++m) s = csub(s, cmul(Lc[i][m], cconj(Lc[k][m])));
                Lc[i][k] = {s.re * inv, s.im * inv};
            }
        }
        // Invert lower triangular Lc -> Li
        cx Li[4][4] = {};
        for (int i = 0; i < 4; ++i) Li[i][i] = {1.f / Lc[i][i].re, 0.f};
        for (int i = 0; i < 4; ++i)
            for (int j = 0; j < i; ++j) {
                cx s = {0.f, 0.f};
                for (int m = j; m < i; ++m) s = cadd(s, cmul(Lc[i][m], Li[m][j]));
                float inv = -1.f / Lc[i][i].re;
                Li[i][j] = {s.re * inv, s.im * inv};
            }
        // Rinv = (Lc^H)^{-1} = Li^H
        for (int i = 0; i < 4; ++i)
            for (int j = 0; j < 4; ++j) {
                rinv[(i * 4 + j) * 2 + 0] = Li[j][i].re;
                rinv[(i * 4 + j) * 2 + 1] = -Li[j][i].im;
            }
    }
}

// ---------------------------------------------------------------------------
// Kernel 2: per-site POVM op + its Cholesky factor L (lower 4x4).
// M_n = (A_t Rinv)^H (A_t Rinv), L_n = chol(M_n). One thread per site.
// ---------------------------------------------------------------------------
__global__ void k_build_L(const float* __restrict__ a_re,
                          const float* __restrict__ a_im,
                          const int* __restrict__ tokens,
                          const float* __restrict__ rinv,
                          float* __restrict__ Lout /* 512*32 floats */) {
    const int n = blockIdx.x * blockDim.x + threadIdx.x;
    if (n >= 512) return;
    const int t = tokens[n];
    cx W[4][4], Ri[4][4], P[4][4], M[4][4];
    #pragma unroll
    for (int i = 0; i < 4; ++i)
        #pragma unroll
        for (int j = 0; j < 4; ++j) {
            W[i][j]  = {a_re[t * 16 + i * 4 + j], a_im[t * 16 + i * 4 + j]};
            Ri[i][j] = {rinv[(i * 4 + j) * 2], rinv[(i * 4 + j) * 2 + 1]};
        }
    #pragma unroll
    for (int i = 0; i < 4; ++i)
        #pragma unroll
        for (int j = 0; j < 4; ++j) {
            cx s = {0.f, 0.f};
            #pragma unroll
            for (int k = 0; k < 4; ++k) s = cadd(s, cmul(W[i][k], Ri[k][j]));
            P[i][j] = s;
        }
    #pragma unroll
    for (int i = 0; i < 4; ++i)
        #pragma unroll
        for (int j = 0; j < 4; ++j) {
            cx s = {0.f, 0.f};
            #pragma unroll
            for (int k = 0; k < 4; ++k) s = cadd(s, cmul(cconj(P[k][i]), P[k][j]));
            M[i][j] = s;
        }
    cx L[4][4] = {};
    for (int k = 0; k < 4; ++k) {
        float d = M[k][k].re;
        for (int m = 0; m < k; ++m) d -= L[k][m].re * L[k][m].re + L[k][m].im * L[k][m].im;
        d = sqrtf(fmaxf(d, 1e-30f));
        L[k][k] = {d, 0.f};
        float inv = 1.f / d;
        for (int i = k + 1; i < 4; ++i) {
            cx s = M[i][k];
            for (int m = 0; m < k; ++m) s = csub(s, cmul(L[i][m], cconj(L[k][m])));
            L[i][k] = {s.re * inv, s.im * inv};
        }
    }
    #pragma unroll
    for (int i = 0; i < 4; ++i)
        #pragma unroll
        for (int j = 0; j < 4; ++j) {
            Lout[n * 32 + (i * 4 + j) * 2 + 0] = L[i][j].re;
            Lout[n * 32 + (i * 4 + j) * 2 + 1] = L[i][j].im;
        }
}

// ---------------------------------------------------------------------------
// Kernel 3: persistent single-workgroup scan over the 512-site chain.
// 256 threads = 8 wave32s.  LDS: S(128KB) + K(128KB) + U 128x32 chunk(32KB).
// Complex GEMM via 4 real v_wmma_f32_16x16x4_f32 streams.
// ---------------------------------------------------------------------------
#define CW 32  // U chunk width (columns)

__global__ __launch_bounds__(256, 1) void k_scan(
    const float* __restrict__ b0_re, const float* __restrict__ b0_im,
    const float* __restrict__ bm_re, const float* __restrict__ bm_im,
    const float* __restrict__ bl_re, const float* __restrict__ bl_im,
    const float* __restrict__ Ls, float* __restrict__ out, int out_size) {
    extern __shared__ float lds[];
    float* Sre = lds;            // 16384
    float* Sim = Sre + 16384;    // 16384
    float* Kre = Sim + 16384;    // 16384
    float* Kim = Kre + 16384;    // 16384
    float* Ure = Kim + 16384;    // 4096 (128 x CW)
    float* Uim = Ure + 4096;     // 4096  -> total 73728 floats = 288 KB

    const int tid  = threadIdx.x;
    const int lane = tid & 31;
    const int wid  = tid >> 5;     // wave id 0..7
    const int m    = lane & 15;    // row/col within 16-tile
    const int kh   = lane >> 4;    // K-pair select (0: k=0,1  1: k=2,3)

    // ---- site 0: S = sum_k v_k^H v_k,  v_k[c] = sum_q conj(L0[q,k]) B0[q,c]
    {
        const float* L0 = Ls;
        for (int idx = tid; idx < 512; idx += 256) {
            const int k = idx >> 7, c = idx & 127;
            float vr = 0.f, vi = 0.f;
            for (int q = k; q < 4; ++q) {
                const float lr = L0[(q * 4 + k) * 2], li = -L0[(q * 4 + k) * 2 + 1];
                const float br = b0_re[q * 128 + c], bb = b0_im[q * 128 + c];
                vr += lr * br - li * bb;
                vi += lr * bb + li * br;
            }
            Ure[idx] = vr; Uim[idx] = vi;
        }
        __syncthreads();
        for (int idx = tid; idx < 16384; idx += 256) {
            const int r = idx >> 7, c = idx & 127;
            float sr = 0.f, si = 0.f;
            #pragma unroll
            for (int k = 0; k < 4; ++k) {
                const float ar = Ure[k * 128 + r], ai = -Uim[k * 128 + r];  // conj(v_k[r])
                const float br = Ure[k * 128 + c], bb = Uim[k * 128 + c];
                sr += ar * br - ai * bb;
                si += ar * bb + ai * br;
            }
            Sre[idx] = sr; Sim[idx] = si;
        }
    }

    // ---- middle sites 1..510 : S' = sum_k K_k^H S K_k
    for (int site = 1; site <= 510; ++site) {
        const size_t bbase = (size_t)(site - 1) * 65536u;
        const float* Lc = Ls + site * 32;
        v8f accRe[8], accIm[8];
        #pragma unroll
        for (int t = 0; t < 8; ++t) { accRe[t] = vzero8(); accIm[t] = vzero8(); }

        for (int kr = 0; kr < 4; ++kr) {
            __syncthreads();  // K reuse safety from previous kraus / site
            // ---- stage K_kr = sum_{q>=kr} conj(L[q,kr]) B_q into LDS
            float cr0[4], ci0[4];
            #pragma unroll
            for (int q = 0; q < 4; ++q) {
                cr0[q] = (q >= kr) ? Lc[(q * 4 + kr) * 2] : 0.f;
                ci0[q] = (q >= kr) ? -Lc[(q * 4 + kr) * 2 + 1] : 0.f;
            }
            for (int idx = tid; idx < 16384; idx += 256) {
                float kre = 0.f, kim = 0.f;
                for (int q = kr; q < 4; ++q) {
                    const float br = bm_re[bbase + (size_t)q * 16384 + idx];
                    const float bb = bm_im[bbase + (size_t)q * 16384 + idx];
                    kre += cr0[q] * br - ci0[q] * bb;
                    kim += cr0[q] * bb + ci0[q] * br;
                }
                Kre[idx] = kre; Kim[idx] = kim;
            }
            // prefetch next site's tensor into cache (global_prefetch_b8)
            if (kr == 0 && site < 510) {
                const float* nr = bm_re + bbase + 65536;
                const float* ni = bm_im + bbase + 65536;
                #pragma unroll
                for (int j = 0; j < 8; ++j) {
                    __builtin_prefetch(nr + (tid * 8 + j) * 32, 0, 1);
                    __builtin_prefetch(ni + (tid * 8 + j) * 32, 0, 1);
                }
            }
            __syncthreads();

            for (int ch = 0; ch < 4; ++ch) {
                const int cbase = ch * CW;
                // ---- GEMM1: U = S @ K[:, cbase:cbase+CW]; wave owns l-strip
                const int l0 = wid * 16;
                #pragma unroll
                for (int ct = 0; ct < 2; ++ct) {
                    const int c0 = cbase + ct * 16;
                    v8f p0 = vzero8(), p1 = vzero8(), p2 = vzero8(), p3 = vzero8();
                    #pragma unroll 4
                    for (int kk = 0; kk < 128; kk += 4) {
                        v2f aR, aI, bR, bI;
                        const int ar = (l0 + m) * 128 + kk + 2 * kh;
                        aR[0] = Sre[ar]; aR[1] = Sre[ar + 1];
                        aI[0] = Sim[ar]; aI[1] = Sim[ar + 1];
                        const int br = (kk + 2 * kh) * 128 + c0 + m;
                        bR[0] = Kre[br]; bR[1] = Kre[br + 128];
                        bI[0] = Kim[br]; bI[1] = Kim[br + 128];
                        p0 = wmma4(aR, bR, p0);
                        p1 = wmma4(aI, bI, p1);
                        p2 = wmma4(aR, bI, p2);
                        p3 = wmma4(aI, bR, p3);
                    }
                    #pragma unroll
                    for (int j = 0; j < 8; ++j) {
                        const int ur = l0 + j + 8 * kh;
                        const int uc = ct * 16 + m;
                        Ure[ur * CW + uc] = p0[j] - p1[j];
                        Uim[ur * CW + uc] = p2[j] + p3[j];
                    }
                }
                __syncthreads();
                // ---- GEMM2: acc[r-strip, cbase:+CW] += K^H U ; wave owns r-strip
                const int r0 = wid * 16;
                #pragma unroll
                for (int ct = 0; ct < 2; ++ct) {
                    const int c0 = ct * 16;
                    v8f p0 = vzero8(), p1 = vzero8(), p2 = vzero8(), p3 = vzero8();
                    #pragma unroll 4
                    for (int kk = 0; kk < 128; kk += 4) {
                        v2f aR, aI, bR, bI;
                        const int ar = (kk + 2 * kh) * 128 + r0 + m;  // K[l, r] (transposed read)
                        aR[0] = Kre[ar]; aR[1] = Kre[ar + 128];
                        aI[0] = Kim[ar]; aI[1] = Kim[ar + 128];
                        const int br = (kk + 2 * kh) * CW + c0 + m;
                        bR[0] = Ure[br]; bR[1] = Ure[br + CW];
                        bI[0] = Uim[br]; bI[1] = Uim[br + CW];
                        p0 = wmma4(aR, bR, p0);
                        p1 = wmma4(aI, bI, p1);
                        p2 = wmma4(aR, bI, p2);
                        p3 = wmma4(aI, bR, p3);
                    }
                    const int t = ch * 2 + ct;
                    accRe[t] += p0 + p1;   // conj(K)*U : re = Kre*Ure + Kim*Uim
                    accIm[t] += p2 - p3;   //             im = Kre*Uim - Kim*Ure
                }
                __syncthreads();
            }
        }
        // ---- write back S' (each wave writes only its own row strip)
        #pragma unroll
        for (int t = 0; t < 8; ++t) {
            #pragma unroll
            for (int j = 0; j < 8; ++j) {
                const int r = wid * 16 + j + 8 * kh;
                const int c = t * 16 + m;
                Sre[r * 128 + c] = accRe[t][j];
                Sim[r * 128 + c] = accIm[t][j];
            }
        }
    }

    // ---- last site: out = sum_k w_k^H S w_k,  w_k = sum_q conj(L[q,k]) Bl[q]
    __syncthreads();
    {
        const float* Lf = Ls + 511 * 32;
        for (int idx = tid; idx < 512; idx += 256) {
            const int k = idx >> 7, r = idx & 127;
            float vr = 0.f, vi = 0.f;
            for (int q = k; q < 4; ++q) {
                const float lr = Lf[(q * 4 + k) * 2], li = -Lf[(q * 4 + k) * 2 + 1];
                const float br = bl_re[q * 128 + r], bb = bl_im[q * 128 + r];
                vr += lr * br - li * bb;
                vi += lr * bb + li * br;
            }
            Ure[idx] = vr; Uim[idx] = vi;
        }
        __syncthreads();
        float pr = 0.f, pi = 0.f;
        for (int idx = tid; idx < 16384; idx += 256) {
            const int r = idx >> 7, c = idx & 127;
            const float sr = Sre[idx], si = Sim[idx];
            #pragma unroll
            for (int k = 0; k < 4; ++k) {
                const float wr_re = Ure[k * 128 + r], wr_im = -Uim[k * 128 + r];
                const float wc_re = Ure[k * 128 + c], wc_im = Uim[k * 128 + c];
                const float tre = wr_re * sr - wr_im * si;
                const float tim = wr_re * si + wr_im * sr;
                pr += tre * wc_re - tim * wc_im;
                pi += tre * wc_im + tim * wc_re;
            }
        }
        Kre[tid] = pr; Kim[tid] = pi;
        __syncthreads();
        for (int s = 128; s > 0; s >>= 1) {
            if (tid < s) { Kre[tid] += Kre[tid + s]; Kim[tid] += Kim[tid + s]; }
            __syncthreads();
        }
        if (tid == 0) {
            out[0] = Kre[0];
            if (out_size > 1) out[1] = Kim[0];
        }
    }
}

// ---------------------------------------------------------------------------
extern "C" void kernel_launch(void* const* d_in, const int* in_sizes, int n_in,
                              void* d_out, int out_size, void* d_ws, size_t ws_size,
                              hipStream_t stream) {
    (void)in_sizes; (void)n_in; (void)ws_size;
    const float* a_re  = (const float*)d_in[0];
    const float* a_im  = (const float*)d_in[1];
    const float* b0_re = (const float*)d_in[2];
    const float* b0_im = (const float*)d_in[3];
    const float* bm_re = (const float*)d_in[4];
    const float* bm_im = (const float*)d_in[5];
    const float* bl_re = (const float*)d_in[6];
    const float* bl_im = (const float*)d_in[7];
    const int*  tokens = (const int*)d_in[8];

    float* ws   = (float*)d_ws;
    float* rinv = ws;        // 32 floats
    float* Ls   = ws + 32;   // 512*32 floats

    k_gram_rinv<<<1, 256, 0, stream>>>(a_re, a_im, rinv);
    k_build_L<<<2, 256, 0, stream>>>(a_re, a_im, tokens, rinv, Ls);

    const size_t ldsBytes = (size_t)(4 * 16384 + 2 * 4096) * sizeof(float);  // 294912
    (void)hipFuncSetAttribute((const void*)k_scan,
                              hipFuncAttributeMaxDynamicSharedMemorySize,
                              (int)ldsBytes);
    k_scan<<<1, 256, ldsBytes, stream>>>(b0_re, b0_im, bm_re, bm_im,
                                         bl_re, bl_im, Ls,
                                         (float*)d_out, out_size);
}